// SimpleEncoder_81312320848131
// MI455X (gfx1250) — compile-verified
//
#include <hip/hip_runtime.h>

// ---------------------------------------------------------------------------
// Problem constants (match reference)
// ---------------------------------------------------------------------------
#define BATCH      1024
#define NPTS       8192
#define CDIM       3
#define PDIM       16
#define KSEL       16
#define DIN        20     // 2 + 2 + 16
#define H          128
#define LOUT       256
#define NTHREADS   256    // threads per block in top-K kernel

typedef __attribute__((ext_vector_type(2))) float v2f;
typedef __attribute__((ext_vector_type(8))) float v8f;

// ---------------------------------------------------------------------------
// Kernel 1: per-batch-row top-16 nearest neighbors + feature construction.
// One block per batch row. Streams 96MB of coords -> bandwidth bound (~4us
// at 23.3 TB/s). Scan uses b128 loads: 4 points (12 floats) = 3 float4s.
// ---------------------------------------------------------------------------
__global__ void __launch_bounds__(NTHREADS)
topk_features_kernel(const float* __restrict__ latent,
                     const float* __restrict__ coords,
                     const float* __restrict__ params,
                     float* __restrict__ X /* [BATCH][DIN] */) {
    const int b = blockIdx.x;
    const int t = threadIdx.x;

    const float lx = latent[0];
    const float ly = latent[1];
    const float* cb = coords + (size_t)b * NPTS * CDIM;

    // ---- per-thread sorted top-16 (register resident: const indices only) --
    float best[KSEL];
    int   bidx[KSEL];
#pragma unroll
    for (int i = 0; i < KSEL; ++i) { best[i] = 3.4e38f; bidx[i] = 0; }

    // 4 points per thread-iteration via three aligned float4 loads:
    // layout [x0 y0 z0 x1 | y1 z1 x2 y2 | z2 x3 y3 z3]
#pragma unroll 2
    for (int j = 0; j < NPTS / (NTHREADS * 4); ++j) {
        const int i0 = (j * NTHREADS + t) * 4;
        const float4* cb4 = reinterpret_cast<const float4*>(cb + (size_t)i0 * 3);
        const float4 f0 = cb4[0];
        const float4 f1 = cb4[1];
        const float4 f2 = cb4[2];

        float px[4], py[4];
        px[0] = f0.x; py[0] = f0.y;
        px[1] = f0.w; py[1] = f1.x;
        px[2] = f1.z; py[2] = f1.w;
        px[3] = f2.y; py[3] = f2.z;

#pragma unroll
        for (int c = 0; c < 4; ++c) {
            const float dx = px[c] - lx;
            const float dy = py[c] - ly;
            const float d  = dx * dx + dy * dy;
            if (d < best[KSEL - 1]) {
                best[KSEL - 1] = d;
                bidx[KSEL - 1] = i0 + c;
#pragma unroll
                for (int s = KSEL - 1; s > 0; --s) {
                    if (best[s] < best[s - 1]) {
                        float tf = best[s]; best[s] = best[s - 1]; best[s - 1] = tf;
                        int   ti = bidx[s]; bidx[s] = bidx[s - 1]; bidx[s - 1] = ti;
                    }
                }
            }
        }
    }

    // ---- cross-thread tree merge in LDS (256 sorted lists -> 1) -----------
    __shared__ float dbuf[NTHREADS * KSEL + (NTHREADS / 2) * KSEL]; // 6144
    __shared__ int   ibuf[NTHREADS * KSEL + (NTHREADS / 2) * KSEL];

    int so = 0;                   // source offset
    int dofs = NTHREADS * KSEL;   // dest offset (4096)
#pragma unroll
    for (int s = 0; s < KSEL; ++s) {
        dbuf[t * KSEL + s] = best[s];
        ibuf[t * KSEL + s] = bidx[s];
    }
    __syncthreads();

    int lists = NTHREADS;
    while (lists > 1) {
        const int half = lists >> 1;
        if (t < half) {
            const int i0 = so + (2 * t) * KSEL;
            const int i1 = so + (2 * t + 1) * KSEL;
            const int o  = dofs + t * KSEL;
            int p = 0, q = 0;
            for (int k = 0; k < KSEL; ++k) {
                const float a = dbuf[i0 + p];
                const float c = dbuf[i1 + q];
                if (a <= c) { dbuf[o + k] = a; ibuf[o + k] = ibuf[i0 + p]; ++p; }
                else        { dbuf[o + k] = c; ibuf[o + k] = ibuf[i1 + q]; ++q; }
            }
        }
        __syncthreads();
        const int tmp = so; so = dofs; dofs = tmp;
        lists = half;
    }
    // final sorted top-16 now lives at ibuf[so .. so+15]

    // ---- gather neighbors, average, emit 20-wide feature row --------------
    __shared__ float gbuf[KSEL][DIN];   // [neighbor][0..2 coords, 3..18 params]
    if (t < KSEL) {
        const int id = ibuf[so + t];
        const float* c3 = coords + ((size_t)b * NPTS + id) * CDIM;
        const float4* p4 = reinterpret_cast<const float4*>(
            params + ((size_t)b * NPTS + id) * PDIM);
        gbuf[t][0] = c3[0];
        gbuf[t][1] = c3[1];
        gbuf[t][2] = c3[2];
#pragma unroll
        for (int s = 0; s < PDIM / 4; ++s) {
            const float4 pv = p4[s];
            gbuf[t][3 + 4 * s + 0] = pv.x;
            gbuf[t][3 + 4 * s + 1] = pv.y;
            gbuf[t][3 + 4 * s + 2] = pv.z;
            gbuf[t][3 + 4 * s + 3] = pv.w;
        }
    }
    __syncthreads();

    if (t < DIN) {
        float v;
        if (t < 2) {
            v = latent[t];
        } else {
            const int f = (t < 4) ? (t - 2) : (3 + (t - 4));
            float sum = 0.f;
#pragma unroll
            for (int s = 0; s < KSEL; ++s) sum += gbuf[s][f];
            v = sum * (1.0f / KSEL);
        }
        X[(size_t)b * DIN + t] = v;
    }
}

// ---------------------------------------------------------------------------
// Kernel 2: fused MLP 20->128 (relu) ->128 (relu) ->256 using
// V_WMMA_F32_16X16X4_F32 (fp32-exact matrix pipe). One wave per 16-row tile,
// 2 waves per block. Hidden activations staged in per-wave LDS tiles.
//
// f32 WMMA fragment layouts (ISA 7.12.2):
//   A 16x4 : lane L holds row m=L&15; v0/v1 = K = (L>=16 ? 2,3 : 0,1)
//   B 4x16 : lane L holds col n=L&15; v0/v1 = K = (L>=16 ? 2,3 : 0,1)
//   C/D    : VGPR r, lanes 0-15 -> (M=r,  N=lane); lanes 16-31 -> (M=r+8)
// ---------------------------------------------------------------------------
#define WPB 2   // waves per block

__global__ void __launch_bounds__(WPB * 32)
mlp_wmma_kernel(const float* __restrict__ X,
                const float* __restrict__ W1, const float* __restrict__ b1,
                const float* __restrict__ W2, const float* __restrict__ b2,
                const float* __restrict__ W3, const float* __restrict__ b3,
                float* __restrict__ out) {
    __shared__ float h1[WPB][16 * H];
    __shared__ float h2[WPB][16 * H];

    const int lane = threadIdx.x & 31;
    const int wv   = threadIdx.x >> 5;
    const int row0 = (blockIdx.x * WPB + wv) * 16;

    const int m  = lane & 15;   // A row / B col / C col
    const int hi = lane >> 4;   // half-wave select
    const int kb = hi * 2;      // K sub-offset inside 4-chunk

    // ---- preload A fragments of X tile (16 x 20, 5 K-chunks) --------------
    v2f aX[5];
#pragma unroll
    for (int kc = 0; kc < 5; ++kc) {
        const float* xr = X + (size_t)(row0 + m) * DIN + kc * 4 + kb;
        aX[kc].x = xr[0];
        aX[kc].y = xr[1];
    }

    // ---- layer 1: X(16x20) @ W1(20x128) + b1, relu -> h1 ------------------
#pragma unroll
    for (int nt = 0; nt < H / 16; ++nt) {
        const int n = nt * 16 + m;
        const float bias = b1[n];
        v8f acc = {bias, bias, bias, bias, bias, bias, bias, bias};
#pragma unroll
        for (int kc = 0; kc < 5; ++kc) {
            v2f bW;
            const float* wr = W1 + (size_t)(kc * 4 + kb) * H + n;
            bW.x = wr[0];
            bW.y = wr[H];
            acc = __builtin_amdgcn_wmma_f32_16x16x4_f32(
                false, aX[kc], false, bW, (short)0, acc, false, false);
        }
#pragma unroll
        for (int r = 0; r < 8; ++r) {
            float v = acc[r];
            v = v > 0.f ? v : 0.f;
            h1[wv][(r + hi * 8) * H + n] = v;
        }
    }
    __syncthreads();

    // ---- layer 2: h1(16x128) @ W2(128x128) + b2, relu -> h2 ---------------
    v2f aH[32];
#pragma unroll
    for (int kc = 0; kc < 32; ++kc) {
        aH[kc].x = h1[wv][m * H + kc * 4 + kb];
        aH[kc].y = h1[wv][m * H + kc * 4 + kb + 1];
    }
#pragma unroll
    for (int nt = 0; nt < H / 16; ++nt) {
        const int n = nt * 16 + m;
        const float bias = b2[n];
        v8f acc = {bias, bias, bias, bias, bias, bias, bias, bias};
#pragma unroll
        for (int kc = 0; kc < 32; ++kc) {
            v2f bW;
            const float* wr = W2 + (size_t)(kc * 4 + kb) * H + n;
            bW.x = wr[0];
            bW.y = wr[H];
            acc = __builtin_amdgcn_wmma_f32_16x16x4_f32(
                false, aH[kc], false, bW, (short)0, acc, false, false);
        }
#pragma unroll
        for (int r = 0; r < 8; ++r) {
            float v = acc[r];
            v = v > 0.f ? v : 0.f;
            h2[wv][(r + hi * 8) * H + n] = v;
        }
    }
    __syncthreads();

    // ---- layer 3: h2(16x128) @ W3(128x256) + b3 -> out --------------------
    v2f aH2[32];
#pragma unroll
    for (int kc = 0; kc < 32; ++kc) {
        aH2[kc].x = h2[wv][m * H + kc * 4 + kb];
        aH2[kc].y = h2[wv][m * H + kc * 4 + kb + 1];
    }
#pragma unroll
    for (int nt = 0; nt < LOUT / 16; ++nt) {
        const int n = nt * 16 + m;
        const float bias = b3[n];
        v8f acc = {bias, bias, bias, bias, bias, bias, bias, bias};
#pragma unroll
        for (int kc = 0; kc < 32; ++kc) {
            v2f bW;
            const float* wr = W3 + (size_t)(kc * 4 + kb) * LOUT + n;
            bW.x = wr[0];
            bW.y = wr[LOUT];
            acc = __builtin_amdgcn_wmma_f32_16x16x4_f32(
                false, aH2[kc], false, bW, (short)0, acc, false, false);
        }
#pragma unroll
        for (int r = 0; r < 8; ++r) {
            out[(size_t)(row0 + r + hi * 8) * LOUT + n] = acc[r];
        }
    }
}

// ---------------------------------------------------------------------------
extern "C" void kernel_launch(void* const* d_in, const int* in_sizes, int n_in,
                              void* d_out, int out_size, void* d_ws, size_t ws_size,
                              hipStream_t stream) {
    const float* latent = (const float*)d_in[0];
    const float* coords = (const float*)d_in[1];
    const float* params = (const float*)d_in[2];
    const float* W1     = (const float*)d_in[3];
    const float* b1     = (const float*)d_in[4];
    const float* W2     = (const float*)d_in[5];
    const float* b2     = (const float*)d_in[6];
    const float* W3     = (const float*)d_in[7];
    const float* b3     = (const float*)d_in[8];

    float* X = (float*)d_ws;   // [BATCH][DIN] = 80 KB scratch

    topk_features_kernel<<<BATCH, NTHREADS, 0, stream>>>(latent, coords, params, X);

    mlp_wmma_kernel<<<BATCH / (16 * WPB), WPB * 32, 0, stream>>>(
        X, W1, b1, W2, b2, W3, b3, (float*)d_out);
}